// GlobalSVD_27419071218247
// MI455X (gfx1250) — compile-verified
//
#include <hip/hip_runtime.h>

// ---------------------------------------------------------------------------
// GlobalSVD on MI455X (gfx1250, wave32).
// 32768 batched 32x32 SVDs via one-sided Jacobi, column-per-lane in VGPRs,
// LDS b128 column exchanges. V is NOT accumulated during iteration; it is
// recovered afterwards as Sigma*Vh = U^T A on the WMMA pipe (sigma cancels
// against d = sigma*dw in the reconstruction). Rank-8 reconstruction also
// runs on v_wmma_f32_16x16x4_f32.
// ---------------------------------------------------------------------------

#define NSWEEPS 8
constexpr int N            = 32;       // H = W = C = 32
constexpr int MODE         = 8;        // retained singular components
constexpr int STAGE_FLOATS = 32 * 36;  // exchange stage; reused as A32/B8/A8
constexpr int AUX_FLOATS   = 16 * 33 + 32;  // Ut[16][33] + sorted sigmas
constexpr int WAVES_PER_BLOCK = 8;

typedef __attribute__((ext_vector_type(2))) float v2f;
typedef __attribute__((ext_vector_type(8))) float v8f;

__device__ __forceinline__ void lds_wait() {
    __builtin_amdgcn_wave_barrier();
    asm volatile("s_wait_dscnt 0x0" ::: "memory");
    __builtin_amdgcn_wave_barrier();
}

// Exchange one 32-float column through the per-wave LDS stage:
// store mine (8 x ds_store_b128), wait, load partner's (8 x ds_load_b128).
// Stride 36 dwords -> all 32 lanes hit distinct bank groups.
__device__ __forceinline__ void exch_col(float* stage, int lane, int partner,
                                         const float* mine, float* theirs) {
    float4* dst = reinterpret_cast<float4*>(stage + lane * 36);
#pragma unroll
    for (int k = 0; k < 8; ++k)
        dst[k] = make_float4(mine[4*k+0], mine[4*k+1], mine[4*k+2], mine[4*k+3]);
    lds_wait();
    const float4* src = reinterpret_cast<const float4*>(stage + partner * 36);
#pragma unroll
    for (int k = 0; k < 8; ++k) {
        float4 v = src[k];
        theirs[4*k+0] = v.x; theirs[4*k+1] = v.y;
        theirs[4*k+2] = v.z; theirs[4*k+3] = v.w;
    }
}

// One-sided Jacobi on the columns of a 32x32 matrix (column-per-lane in VGPRs).
// On exit: col = U column (normalized), sigma = singular value,
// rank = descending rank (0 = largest) among the 32 columns.
__device__ void jacobi_u(float col[N], float* stage, int lane,
                         float& sigma_out, int& rank_out) {
    float other[N];
    for (int sweep = 0; sweep < NSWEEPS; ++sweep) {
        for (int r = 0; r < N - 1; ++r) {
            // circle-method round-robin pairing: lane 31 fixed, others rotate
            int partner;
            if (lane == N - 1)      partner = r;
            else if (lane == r)     partner = N - 1;
            else                    partner = (2 * r - lane + 62) % 31;

            exch_col(stage, lane, partner, col, other);

            // All three Gram terms from registers (no shfl in the chain).
            float self = 0.0f, oself = 0.0f, cross = 0.0f;
#pragma unroll
            for (int i = 0; i < N; ++i) {
                self  += col[i]   * col[i];
                oself += other[i] * other[i];
                cross += col[i]   * other[i];
            }
            bool  isp = lane < partner;           // this lane plays role p
            float app = isp ? self  : oself;
            float aqq = isp ? oself : self;
            float apq = cross;                    // symmetric: both lanes agree

            float t = 0.0f;
            if (fabsf(apq) > 1e-30f) {
                float tau = (aqq - app) / (2.0f * apq);
                t = (tau >= 0.0f ? 1.0f : -1.0f) /
                    (fabsf(tau) + sqrtf(1.0f + tau * tau));
            }
            float cs = 1.0f / sqrtf(1.0f + t * t);
            float sn = cs * t;
            float sl = isp ? -sn : sn;   // p: c*a_p - s*a_q ; q: s*a_p + c*a_q
#pragma unroll
            for (int i = 0; i < N; ++i) col[i] = cs * col[i] + sl * other[i];
        }
    }

    float nrm2 = 0.0f;
#pragma unroll
    for (int i = 0; i < N; ++i) nrm2 += col[i] * col[i];
    float sigma = sqrtf(nrm2);
    float inv   = (sigma > 0.0f) ? (1.0f / sigma) : 0.0f;
#pragma unroll
    for (int i = 0; i < N; ++i) col[i] *= inv;

    // descending rank via LDS broadcast of all 32 sigmas
    stage[lane] = sigma;
    lds_wait();
    int rank = 0;
#pragma unroll
    for (int j = 0; j < N; ++j) {
        float sj = stage[j];
        rank += (sj > sigma) || (sj == sigma && j < lane) ? 1 : 0;
    }
    lds_wait();  // all sigma reads done before the stage is reused
    sigma_out = sigma;
    rank_out  = rank;
}

// rawV16 = Ut(16x32) @ A32(32x32) on the matrix pipe.
// Result tiles: vacc[tj], VGPR r, lanes 0..15 -> (m=r, w=tj*16+lane).
__device__ __forceinline__ void wmma_ut_a(const float* Ut, const float* A32,
                                          int lane, v8f vacc[2]) {
    const int hl = lane & 15;
    const int kh = (lane >> 4) << 1;   // K sub-offset 0 / 2
#pragma unroll
    for (int tj = 0; tj < 2; ++tj) {
        v8f acc = {};
#pragma unroll
        for (int kc = 0; kc < N; kc += 4) {
            v2f a, b;
            a.x = Ut[hl * 33 + kc + kh + 0];
            a.y = Ut[hl * 33 + kc + kh + 1];
            b.x = A32[(kc + kh + 0) * N + tj * 16 + hl];
            b.y = A32[(kc + kh + 1) * N + tj * 16 + hl];
            acc = __builtin_amdgcn_wmma_f32_16x16x4_f32(
                      false, a, false, b, (short)0, acc, false, false);
        }
        vacc[tj] = acc;
    }
}

// ---------------------------------------------------------------------------
// Kernel 1: SVD of the 32 weight matrices w[c]; top-8 U, sigma, and
// normalized Vh (recovered as (U^T A)/sigma via WMMA) -> workspace.
// ---------------------------------------------------------------------------
__global__ __launch_bounds__(256) void svd_w_kernel(const float* __restrict__ w,
                                                    float* __restrict__ wU,
                                                    float* __restrict__ wV,
                                                    float* __restrict__ wd) {
    __shared__ float lds[WAVES_PER_BLOCK][STAGE_FLOATS + AUX_FLOATS];
    const int lane = threadIdx.x & 31;
    const int wv   = threadIdx.x >> 5;
    float* stage   = lds[wv];
    float* aux     = stage + STAGE_FLOATS;
    const int c    = blockIdx.x * WAVES_PER_BLOCK + wv;

    float a0[N], col[N];
#pragma unroll
    for (int i = 0; i < N; ++i) { a0[i] = w[c * N * N + i * N + lane]; col[i] = a0[i]; }

    float sigma; int rank;
    jacobi_u(col, stage, lane, sigma, rank);

    if (rank < MODE) {
        wd[c * MODE + rank] = sigma;
#pragma unroll
        for (int i = 0; i < N; ++i) wU[c * N * MODE + i * MODE + rank] = col[i];
    }

    // Stage operands for V = (U^T A) / sigma
    float* A32     = stage;          // [32][32]
    float* Ut      = aux;            // [16][33]  rows 8..15 zeroed
    float* sortSig = aux + 16 * 33;  // sigma sorted by rank
#pragma unroll
    for (int i = 0; i < N; ++i) A32[i * N + lane] = a0[i];
    if (rank < 16) {
        sortSig[rank] = sigma;
#pragma unroll
        for (int i = 0; i < N; ++i) Ut[rank * 33 + i] = (rank < MODE) ? col[i] : 0.0f;
    }
    lds_wait();

    v8f vacc[2];
    wmma_ut_a(Ut, A32, lane, vacc);

    if (lane < 16) {
#pragma unroll
        for (int tj = 0; tj < 2; ++tj) {
#pragma unroll
            for (int r = 0; r < MODE; ++r) {
                float s   = sortSig[r];
                float inv = (s > 0.0f) ? (1.0f / s) : 0.0f;
                wV[c * MODE * N + r * N + tj * 16 + lane] = vacc[tj][r] * inv;
            }
        }
    }
}

// ---------------------------------------------------------------------------
// Kernel 2: per-(b,t,c) SVD + gated rank-8 reconstruction.
// Block = 256 threads = 8 waves = 8 consecutive channels of one (b,t).
//   rawV = U^T A  (16 WMMA)   [sigma cancels against d = sigma*dw]
//   out  = (Uq.*Uw.*dw) @ (rawV.*Vw)   (8 WMMA)
// ---------------------------------------------------------------------------
__global__ __launch_bounds__(256) void svd_mix_kernel(const float* __restrict__ x,
                                                      const float* __restrict__ wU,
                                                      const float* __restrict__ wV,
                                                      const float* __restrict__ wd,
                                                      float* __restrict__ out) {
    __shared__ float lds[WAVES_PER_BLOCK][STAGE_FLOATS + AUX_FLOATS];
    const int lane = threadIdx.x & 31;
    const int wv   = threadIdx.x >> 5;
    float* stage   = lds[wv];
    float* aux     = stage + STAGE_FLOATS;

    const int bt = blockIdx.x >> 2;                 // (b*32 + t)
    const int c  = ((blockIdx.x & 3) << 3) + wv;    // channel

    // xt[b,t,c][h][w] = x[b,t,h,w,c]; lane owns column w = lane.
    const float* xbt = x + (size_t)bt * (N * N * N);
    float a0[N], col[N];
#pragma unroll
    for (int i = 0; i < N; ++i) {
        a0[i]  = xbt[(size_t)(i * N + lane) * N + c];
        col[i] = a0[i];
    }

    float sigma; int rank;
    jacobi_u(col, stage, lane, sigma, rank);
    (void)sigma;  // cancels: d_m * Vh_m = (sigma*dw_m) * rawV_m / sigma

    // Stage operands for rawV = U^T A
    float* A32 = stage;   // [32][32]
    float* Ut  = aux;     // [16][33], rows 8..15 zero
#pragma unroll
    for (int i = 0; i < N; ++i) A32[i * N + lane] = a0[i];
    if (rank < 16) {
#pragma unroll
        for (int i = 0; i < N; ++i) Ut[rank * 33 + i] = (rank < MODE) ? col[i] : 0.0f;
    }
    lds_wait();

    v8f vacc[2];
    wmma_ut_a(Ut, A32, lane, vacc);
    lds_wait();   // all A32 fragment reads retired before overwriting stage

    // B8[m][w] = rawV[m][w] * Vw-gate ; A8[h][m] = U[h,m] * Uw-gate * dw_m
    float* B8 = stage;         // [8][32]
    float* A8 = stage + 256;   // [32][8]
    if (lane < 16) {
#pragma unroll
        for (int tj = 0; tj < 2; ++tj) {
#pragma unroll
            for (int r = 0; r < MODE; ++r) {
                int wcol = tj * 16 + lane;
                B8[r * N + wcol] = vacc[tj][r] * wV[c * MODE * N + r * N + wcol];
            }
        }
    }
    if (rank < MODE) {
        float dwm = wd[c * MODE + rank];
#pragma unroll
        for (int i = 0; i < N; ++i)
            A8[i * MODE + rank] = col[i] * wU[c * N * MODE + i * MODE + rank] * dwm;
    }
    lds_wait();

    // out(32x32) = A8 @ B8 : four 16x16 tiles, K=8 as two 16x16x4 steps.
    const int hl = lane & 15;
    const int kh = (lane >> 4) << 1;   // K sub-offset 0 / 2
    const int mh = (lane >> 4) << 3;   // D row offset 0 / 8
#pragma unroll
    for (int ti = 0; ti < 2; ++ti) {
#pragma unroll
        for (int tj = 0; tj < 2; ++tj) {
            v8f acc = {};
#pragma unroll
            for (int kc = 0; kc < MODE; kc += 4) {
                v2f a, b;
                a.x = A8[(ti * 16 + hl) * MODE + kc + kh + 0];
                a.y = A8[(ti * 16 + hl) * MODE + kc + kh + 1];
                b.x = B8[(kc + kh + 0) * N + tj * 16 + hl];
                b.y = B8[(kc + kh + 1) * N + tj * 16 + hl];
                acc = __builtin_amdgcn_wmma_f32_16x16x4_f32(
                          false, a, false, b, (short)0, acc, false, false);
            }
            const int wcol = tj * 16 + hl;
#pragma unroll
            for (int rrow = 0; rrow < 8; ++rrow) {
                const int h = ti * 16 + mh + rrow;
                out[((size_t)bt * (N * N) + h * N + wcol) * N + c] = acc[rrow];
            }
        }
    }
}

extern "C" void kernel_launch(void* const* d_in, const int* in_sizes, int n_in,
                              void* d_out, int out_size, void* d_ws, size_t ws_size,
                              hipStream_t stream) {
    const float* x = (const float*)d_in[0];   // [32,32,32,32,32]
    const float* w = (const float*)d_in[1];   // [32,32,32]
    float* out = (float*)d_out;               // [32,32,32,32,32]

    float* wU = (float*)d_ws;                 // 32*32*8 floats
    float* wV = wU + 32 * N * MODE;           // 32*8*32 floats
    float* wd = wV + 32 * MODE * N;           // 32*8   floats

    // 32 weight SVDs: 4 blocks x 8 waves
    svd_w_kernel<<<4, 256, 0, stream>>>(w, wU, wV, wd);
    // 32768 data SVDs + reconstruction: (b,t) x 4 channel-groups
    svd_mix_kernel<<<32 * 32 * 4, 256, 0, stream>>>(x, wU, wV, wd, out);
}